// LindeBuzoGrayAlgorithm_30399778521128
// MI455X (gfx1250) — compile-verified
//
#include <hip/hip_runtime.h>

#define T_PTS 65536
#define MDIM  64
#define KCB   256
#define BIGV  1e10f
#define PERT  1e-5f

typedef __attribute__((ext_vector_type(2))) float v2f;
typedef __attribute__((ext_vector_type(8))) float v8f;

__device__ __forceinline__ float hrand(unsigned a, unsigned b, unsigned c) {
  unsigned h = a * 0x9E3779B1u + b * 0x85EBCA77u + c * 0xC2B2AE3Du;
  h ^= h >> 16; h *= 0x7FEB352Du; h ^= h >> 15; h *= 0x846CA68Bu; h ^= h >> 16;
  return (float)(int)h * 4.6566129e-10f;  // uniform [-1,1)
}

__global__ void lbg_zero(float* __restrict__ p, int n) {
  int i = blockIdx.x * blockDim.x + threadIdx.x;
  if (i < n) p[i] = 0.f;
}

// Column sums of x -> acc[64] (for mean init). 256 blocks x 256 thr = 1024 row groups.
__global__ void lbg_colsum(const float* __restrict__ x, float* __restrict__ acc) {
  int c = threadIdx.x & 63;
  int g = blockIdx.x * 4 + (threadIdx.x >> 6);
  float s = 0.f;
  for (int r = g; r < T_PTS; r += 1024) s += x[(size_t)r * MDIM + c];
  atomicAdd(&acc[c], s);
}

__global__ void lbg_initcb(float* __restrict__ cb, const float* __restrict__ acc) {
  int i = blockIdx.x * blockDim.x + threadIdx.x;  // 16384 threads
  int k = i >> 6, d = i & 63;
  cb[i] = (k == 0) ? acc[d] * (1.f / T_PTS) : BIGV;
}

__global__ void lbg_split(float* __restrict__ cb, int curr, unsigned seed) {
  int k = blockIdx.x, d = threadIdx.x;
  float r  = PERT * hrand(seed, (unsigned)k, (unsigned)d);
  float c0 = cb[k * MDIM + d];
  cb[(curr + k) * MDIM + d] = c0 - r;
  cb[k * MDIM + d]          = c0 + r;
}

// Precompute ||c_k||^2 once per e-step (removes per-wave recomputation).
__global__ void lbg_csq(const float* __restrict__ cb, float* __restrict__ csq) {
  int k = threadIdx.x;  // 256 threads, 1 block
  const float* cr = cb + k * MDIM;
  float s = 0.f;
  #pragma unroll 8
  for (int d = 0; d < MDIM; ++d) s = fmaf(cr[d], cr[d], s);
  csq[k] = s;
}

// E-step: per wave, TWO 16-row tiles of x vs ntiles*16 codebook columns via
// V_WMMA_F32_16X16X4_F32. B fragments batched per tile (one clause, one wait),
// then 32 back-to-back WMMAs; score = csq - 2 x.c; 16-lane shfl_xor butterfly
// argmin per row. Writes idx (as float); optionally accumulates distortion.
__global__ void __launch_bounds__(256) lbg_estep(
    const float* __restrict__ x, const float* __restrict__ cb,
    const float* __restrict__ csqv,
    float* __restrict__ idxf, float* __restrict__ dsum, int ntiles) {
  const int lane = threadIdx.x & 31;
  const int wave = threadIdx.x >> 5;
  const int rowBase = (blockIdx.x * 8 + wave) * 32;
  const int lm = lane & 15;
  const int hi = lane >> 4;

  // A fragments: lane holds row lm (and lm+16), elems 4*kc + 2*hi + {0,1}
  v2f a0[16], a1[16];
  const float* xr0 = x + (size_t)(rowBase + lm) * MDIM;
  const float* xr1 = x + (size_t)(rowBase + 16 + lm) * MDIM;
  #pragma unroll
  for (int kc = 0; kc < 16; ++kc) {
    a0[kc] = *(const v2f*)(xr0 + kc * 4 + 2 * hi);
    a1[kc] = *(const v2f*)(xr1 + kc * 4 + 2 * hi);
  }
  float xsq0 = 0.f, xsq1 = 0.f;
  #pragma unroll
  for (int kc = 0; kc < 16; ++kc) {
    xsq0 += a0[kc].x * a0[kc].x + a0[kc].y * a0[kc].y;
    xsq1 += a1[kc].x * a1[kc].x + a1[kc].y * a1[kc].y;
  }
  xsq0 += __shfl_xor(xsq0, 16, 32);  // full ||x_row||^2 in all lanes
  xsq1 += __shfl_xor(xsq1, 16, 32);

  float bS0[8], bS1[8]; int bC0[8], bC1[8];
  #pragma unroll
  for (int v = 0; v < 8; ++v) { bS0[v] = bS1[v] = 3.0e38f; bC0[v] = bC1[v] = 0; }

  for (int t = 0; t < ntiles; ++t) {
    const int col = t * 16 + lm;
    const float* cr = cb + (size_t)col * MDIM;
    v2f b[16];
    #pragma unroll
    for (int kc = 0; kc < 16; ++kc)        // one load clause, single wait
      b[kc] = *(const v2f*)(cr + kc * 4 + 2 * hi);
    const float csq = csqv[col];
    v8f acc0 = {}, acc1 = {};
    #pragma unroll
    for (int kc = 0; kc < 16; ++kc) {      // dense WMMA chain, B reused 2x
      acc0 = __builtin_amdgcn_wmma_f32_16x16x4_f32(
          false, a0[kc], false, b[kc], (short)0, acc0, false, false);
      acc1 = __builtin_amdgcn_wmma_f32_16x16x4_f32(
          false, a1[kc], false, b[kc], (short)0, acc1, false, false);
    }
    #pragma unroll
    for (int v = 0; v < 8; ++v) {          // D element (M = v + 8*hi, N = lm)
      float s0 = fmaf(-2.f, acc0[v], csq);
      float s1 = fmaf(-2.f, acc1[v], csq);
      if (s0 < bS0[v]) { bS0[v] = s0; bC0[v] = col; }
      if (s1 < bS1[v]) { bS1[v] = s1; bC1[v] = col; }
    }
  }

  // Reduce argmin across the 16 lanes of each row group (xor<16 stays in half).
  #pragma unroll
  for (int v = 0; v < 8; ++v) {
    float s0 = bS0[v], s1 = bS1[v]; int c0 = bC0[v], c1 = bC1[v];
    #pragma unroll
    for (int off = 1; off < 16; off <<= 1) {
      float t0 = __shfl_xor(s0, off, 32); int d0 = __shfl_xor(c0, off, 32);
      float t1 = __shfl_xor(s1, off, 32); int d1 = __shfl_xor(c1, off, 32);
      if (t0 < s0 || (t0 == s0 && d0 < c0)) { s0 = t0; c0 = d0; }
      if (t1 < s1 || (t1 == s1 && d1 < c1)) { s1 = t1; c1 = d1; }
    }
    bS0[v] = s0; bC0[v] = c0; bS1[v] = s1; bC1[v] = c1;
  }

  float wS0 = 0.f, wS1 = 0.f; int wC0 = 0, wC1 = 0;
  #pragma unroll
  for (int v = 0; v < 8; ++v)
    if (lm == v) { wS0 = bS0[v]; wC0 = bC0[v]; wS1 = bS1[v]; wC1 = bC1[v]; }
  float xn0 = __shfl(xsq0, lm + 8 * hi, 32);
  float xn1 = __shfl(xsq1, lm + 8 * hi, 32);
  bool writer = (lm < 8);
  if (writer) {
    idxf[rowBase + lm + 8 * hi]      = (float)wC0;
    idxf[rowBase + 16 + lm + 8 * hi] = (float)wC1;
  }
  if (dsum != nullptr) {
    float contrib = writer ? (xn0 + wS0 + xn1 + wS1) : 0.f;
    #pragma unroll
    for (int off = 16; off >= 1; off >>= 1) contrib += __shfl_xor(contrib, off, 32);
    if (lane == 0) atomicAdd(dsum, contrib);
  }
}

// Segment sums: wave-cooperative LDS accumulation, two 128-cluster halves.
__global__ void __launch_bounds__(256) lbg_segsum(
    const float* __restrict__ x, const float* __restrict__ idxf,
    float* __restrict__ cnt, float* __restrict__ sums) {
  __shared__ float scnt[128];
  __shared__ float ssum[128 * 64];
  const int lane = threadIdx.x & 31;
  const int wave = threadIdx.x >> 5;
  const int perBlock = T_PTS / gridDim.x;
  const int start = blockIdx.x * perBlock;
  for (int h = 0; h < 2; ++h) {
    const int lo = h * 128;
    for (int i = threadIdx.x; i < 128; i += blockDim.x) scnt[i] = 0.f;
    for (int i = threadIdx.x; i < 128 * 64; i += blockDim.x) ssum[i] = 0.f;
    __syncthreads();
    for (int p = wave; p < perBlock; p += 8) {
      const int i = start + p;
      const int j = (int)idxf[i];
      if (j >= lo && j < lo + 128) {
        const int jl = j - lo;
        const float2 v = ((const float2*)(x + (size_t)i * MDIM))[lane];
        atomicAdd(&ssum[jl * 64 + 2 * lane],     v.x);  // 64 banks, conflict-free
        atomicAdd(&ssum[jl * 64 + 2 * lane + 1], v.y);
        if (lane == 0) atomicAdd(&scnt[jl], 1.f);
      }
    }
    __syncthreads();
    for (int i = threadIdx.x; i < 128; i += blockDim.x)
      if (scnt[i] != 0.f) atomicAdd(&cnt[lo + i], scnt[i]);
    for (int i = threadIdx.x; i < 128 * 64; i += blockDim.x)
      if (ssum[i] != 0.f) atomicAdd(&sums[lo * 64 + i], ssum[i]);
    __syncthreads();
  }
}

// Centroid update incl. empty-cluster handling (single block).
__global__ void __launch_bounds__(256) lbg_update(
    float* __restrict__ cb, const float* __restrict__ cnt,
    const float* __restrict__ sums, int curr, unsigned seed) {
  __shared__ float sn[256];
  __shared__ float centm[64];
  __shared__ float rmean[64];
  __shared__ int sm, sempty;
  const int t = threadIdx.x;
  sn[t] = (t < curr) ? cnt[t] : -1.f;
  if (t < 64) rmean[t] = 0.f;
  if (t == 0) sempty = 0;
  __syncthreads();
  if (t == 0) {  // argmax (first max wins, like jnp.argmax)
    int m = 0; float best = sn[0];
    for (int k = 1; k < curr; ++k) if (sn[k] > best) { best = sn[k]; m = k; }
    sm = m;
  }
  __syncthreads();
  const int m = sm;
  if (t < 64) centm[t] = sums[m * 64 + t] / fmaxf(sn[m], 1.f);
  __syncthreads();
  if (t < curr) {
    const float n = sn[t];
    if (n >= 1.f) {  // MIN_DATA
      if (t != m)
        for (int d = 0; d < 64; ++d) cb[t * 64 + d] = sums[t * 64 + d] / n;
    } else {
      atomicAdd(&sempty, 1);
      for (int d = 0; d < 64; ++d) {
        float r = PERT * hrand(seed, (unsigned)t, (unsigned)d);
        cb[t * 64 + d] = centm[d] - r;
        atomicAdd(&rmean[d], r);
      }
    }
  }
  __syncthreads();
  if (t < 64) {
    float add = (sempty > 0) ? rmean[t] / (float)sempty : 0.f;
    cb[m * 64 + t] = centm[t] + add;
  }
}

__global__ void lbg_writedist(float* __restrict__ dist, const float* __restrict__ dsum) {
  dist[0] = dsum[0] * (1.f / T_PTS);
}

extern "C" void kernel_launch(void* const* d_in, const int* in_sizes, int n_in,
                              void* d_out, int out_size, void* d_ws, size_t ws_size,
                              hipStream_t stream) {
  (void)in_sizes; (void)n_in; (void)out_size; (void)ws_size;
  const float* x = (const float*)d_in[0];
  float* out  = (float*)d_out;
  float* cb   = out;                       // 256*64  (output cb used as working buffer)
  float* idxf = out + KCB * MDIM;          // 65536   (indices as float)
  float* dist = out + KCB * MDIM + T_PTS;  // 1
  float* ws   = (float*)d_ws;
  float* dsum = ws;                        // [0]
  float* macc = ws + 64;                   // [64..128)    column accumulators
  float* cnt  = ws + 128;                  // [128..384)   cluster counts
  float* sums = ws + 384;                  // [384..16768) cluster sums
  float* csqv = ws + 16768;                // [16768..17024) ||c||^2 per cluster

  // init = mean
  lbg_zero<<<1, 64, 0, stream>>>(macc, 64);
  lbg_colsum<<<256, 256, 0, stream>>>(x, macc);
  lbg_initcb<<<64, 256, 0, stream>>>(cb, macc);

  int curr = 1;
  for (int level = 0; level < 8; ++level) {
    const int curr2 = curr * 2;                 // 2,4,...,256
    lbg_split<<<curr, 64, 0, stream>>>(cb, curr, 1000u + (unsigned)level);
    int ntiles = (curr2 + 15) / 16; if (ntiles < 1) ntiles = 1;
    for (int it = 0; it < 4; ++it) {
      lbg_zero<<<66, 256, 0, stream>>>(ws, 16768);  // dsum + cnt + sums
      lbg_csq<<<1, 256, 0, stream>>>(cb, csqv);
      lbg_estep<<<256, 256, 0, stream>>>(x, cb, csqv, idxf, dsum, ntiles);
      lbg_segsum<<<128, 256, 0, stream>>>(x, idxf, cnt, sums);
      lbg_update<<<1, 256, 0, stream>>>(cb, cnt, sums, curr2,
                                        7000u + (unsigned)(level * 16 + it));
    }
    curr = curr2;
  }
  lbg_writedist<<<1, 1, 0, stream>>>(dist, dsum);  // distortion of last e-step
  lbg_csq<<<1, 256, 0, stream>>>(cb, csqv);
  lbg_estep<<<256, 256, 0, stream>>>(x, cb, csqv, idxf, nullptr, 16);  // final assignment
}